// Orthonorm_6227702579715
// MI455X (gfx1250) — compile-verified
//
#include <hip/hip_runtime.h>

typedef float v2f __attribute__((ext_vector_type(2)));
typedef float v8f __attribute__((ext_vector_type(8)));
typedef int   gv4i __attribute__((vector_size(4 * sizeof(int))));  // matches builtin param

#define UDIM 64
#define XPAD 68   // padded LDS row stride in floats: 68*4 B keeps 16B alignment for b128;
                  // bank = (4*row + col) mod 64 -> conflict-free fragment reads

// ---------------------------------------------------------------------------
// CDNA5 async global->LDS copy of 16 bytes per lane (GLOBAL_LOAD_ASYNC_TO_LDS_B128,
// tracked by ASYNCcnt). Builtin signature recovered from clang diagnostic:
// param0 = AS(1) gv4i* (global source). Dest assumed AS(3) gv4i*.
__device__ __forceinline__ void async_copy_b128(const float* gsrc, float* ldst) {
#if __has_builtin(__builtin_amdgcn_global_load_async_to_lds_b128)
    __builtin_amdgcn_global_load_async_to_lds_b128(
        (__attribute__((address_space(1))) gv4i*)(uintptr_t)gsrc,
        (__attribute__((address_space(3))) gv4i*)(uintptr_t)(unsigned int)(uintptr_t)ldst,
        0, 0);
#else
    unsigned int l = (unsigned int)(uintptr_t)ldst;
    asm volatile("global_load_async_to_lds_b128 %0, %1, off"
                 :: "v"(l), "v"((unsigned long long)(uintptr_t)gsrc)
                 : "memory");
#endif
}

__device__ __forceinline__ void wait_async0() {
#if __has_builtin(__builtin_amdgcn_s_wait_asynccnt)
    __builtin_amdgcn_s_wait_asynccnt(0);
#else
    asm volatile("s_wait_asynccnt 0x0" ::: "memory");
#endif
}

// ---------------------------------------------------------------------------
__global__ void zero_kernel(float* __restrict__ p, int n) {
    int i = blockIdx.x * blockDim.x + threadIdx.x;
    if (i < n) p[i] = 0.0f;
}

// ---------------------------------------------------------------------------
// Gram: g[i][j] = sum_n x[n][i]*x[n][j] via V_WMMA_F32_16X16X4_F32.
// 512 threads = 16 waves; wave w owns tile (i0=(w>>2)*16, j0=(w&3)*16).
// Double-buffered LDS chunks (64 rows), async global->LDS, 1 barrier/chunk.
__global__ __launch_bounds__(512) void gram_kernel(const float* __restrict__ x,
                                                   float* __restrict__ gram,
                                                   int nChunks) {
    __shared__ float sx[2][UDIM * XPAD];

    const int tid  = threadIdx.x;
    const int wave = tid >> 5;
    const int lane = tid & 31;
    const int half = lane >> 4;         // A/B 16x4 f32 layout: half selects K pair
    const int l    = lane & 15;
    const int ti   = (wave >> 2) << 4;
    const int tj   = (wave & 3) << 4;

    // per-thread staging slots (two float4 per thread per chunk)
    const int i0 = tid, i1 = tid + 512;
    float* d0 = &sx[0][(i0 >> 4) * XPAD + (i0 & 15) * 4];
    float* d1 = &sx[0][(i1 >> 4) * XPAD + (i1 & 15) * 4];
    const int bufStride = UDIM * XPAD;

    v8f c = {};

    int chunk = blockIdx.x;
    int p = 0;
    if (chunk < nChunks) {
        const float* src = x + (size_t)chunk * (UDIM * UDIM);
        async_copy_b128(src + i0 * 4, d0);
        async_copy_b128(src + i1 * 4, d1);
    }

    for (; chunk < nChunks; chunk += gridDim.x) {
        wait_async0();        // this wave's async loads into buf[p] done
        __syncthreads();      // all waves' loads done AND prior compute finished

        int next = chunk + gridDim.x;
        if (next < nChunks) { // prefetch next chunk into buf[p^1] during compute
            const float* src = x + (size_t)next * (UDIM * UDIM);
            float* dst0 = d0 + (p ^ 1) * bufStride;
            float* dst1 = d1 + (p ^ 1) * bufStride;
            async_copy_b128(src + i0 * 4, dst0);
            async_copy_b128(src + i1 * 4, dst1);
        }

        const float* buf = sx[p];
#pragma unroll
        for (int k = 0; k < UDIM; k += 4) {
            v2f a, b;
            // A[m][kk] = x[n0+kk][ti+m]  (x^T tile)
            a.x = buf[(k + 2 * half)     * XPAD + ti + l];
            a.y = buf[(k + 2 * half + 1) * XPAD + ti + l];
            // B[kk][j] = x[n0+kk][tj+j]
            b.x = buf[(k + 2 * half)     * XPAD + tj + l];
            b.y = buf[(k + 2 * half + 1) * XPAD + tj + l];
            c = __builtin_amdgcn_wmma_f32_16x16x4_f32(false, a, false, b,
                                                      (short)0, c, false, false);
        }
        p ^= 1;
    }

    // C/D layout: VGPR r -> lanes 0-15: (M=r, N=l), lanes 16-31: (M=r+8, N=l)
#pragma unroll
    for (int r = 0; r < 8; ++r) {
        int m = r + 8 * half;
        atomicAdd(&gram[(ti + m) * UDIM + (tj + l)], c[r]);
    }
}

// ---------------------------------------------------------------------------
// Cholesky of (gram + eps*I), then W = inv(L)^T * sqrt(64). One block, 64 threads.
__global__ __launch_bounds__(64) void factor_kernel(const float* __restrict__ gram,
                                                    float* __restrict__ W) {
    __shared__ float sL[UDIM * UDIM];
    __shared__ float sI[UDIM * UDIM];
    const int t = threadIdx.x;

    for (int i = t; i < UDIM * UDIM; i += UDIM) {
        float v = gram[i];
        int r = i >> 6, cc = i & 63;
        if (r == cc) v += 1e-6f;
        sL[i] = v;
    }
    __syncthreads();

    for (int k = 0; k < UDIM; ++k) {
        if (t == 0) sL[k * UDIM + k] = sqrtf(sL[k * UDIM + k]);
        __syncthreads();
        if (t > k) sL[t * UDIM + k] /= sL[k * UDIM + k];
        __syncthreads();
        if (t > k) {
            float lik = sL[t * UDIM + k];
            for (int j = k + 1; j <= t; ++j)
                sL[t * UDIM + j] -= lik * sL[j * UDIM + k];
        }
        __syncthreads();
    }

    for (int i = 0; i < UDIM; ++i) {
        float acc = (i == t) ? 1.0f : 0.0f;
        for (int j = t; j < i; ++j) acc -= sL[i * UDIM + j] * sI[j * UDIM + t];
        sI[i * UDIM + t] = (i >= t) ? acc / sL[i * UDIM + i] : 0.0f;
    }
    __syncthreads();

    for (int idx = t; idx < UDIM * UDIM; idx += UDIM) {
        int i = idx >> 6, j = idx & 63;
        W[idx] = sI[j * UDIM + i] * 8.0f;   // sqrt(64)
    }
}

// ---------------------------------------------------------------------------
// out = x @ W. 256 threads = 8 waves; wave w owns one 16-row tile (4 WMMA accums).
__global__ __launch_bounds__(256) void apply_kernel(const float* __restrict__ x,
                                                    const float* __restrict__ W,
                                                    float* __restrict__ out,
                                                    int nRowTiles) {
    __shared__ float sW[UDIM * XPAD];
    __shared__ float sX[128 * XPAD];

    const int tid  = threadIdx.x;
    const int wave = tid >> 5;
    const int lane = tid & 31;
    const int half = lane >> 4;
    const int l    = lane & 15;

    const int tile0    = blockIdx.x * 8;
    const int rowsBase = tile0 * 16;
    const int nRows    = min(128, nRowTiles * 16 - rowsBase);

    // async-stage W (1024 float4) and x rows (nRows*16 float4)
    for (int i = tid; i < (UDIM * UDIM) / 4; i += 256)
        async_copy_b128(W + i * 4, &sW[(i >> 4) * XPAD + (i & 15) * 4]);
    {
        const float* xb = x + (size_t)rowsBase * UDIM;
        int n4 = nRows * 16;
        for (int i = tid; i < n4; i += 256)
            async_copy_b128(xb + i * 4, &sX[(i >> 4) * XPAD + (i & 15) * 4]);
    }
    wait_async0();
    __syncthreads();

    if (tile0 + wave < nRowTiles) {   // wave-uniform: EXEC all-ones inside
        v8f c0 = {}, c1 = {}, c2 = {}, c3 = {};
        const int rbase = wave * 16;

#pragma unroll
        for (int k = 0; k < UDIM; k += 4) {
            v2f a;
            a.x = sX[(rbase + l) * XPAD + k + 2 * half];
            a.y = sX[(rbase + l) * XPAD + k + 2 * half + 1];
            const int r0 = (k + 2 * half) * XPAD;
            const int r1 = (k + 2 * half + 1) * XPAD;
            v2f b;
            b.x = sW[r0 + 0  + l]; b.y = sW[r1 + 0  + l];
            c0 = __builtin_amdgcn_wmma_f32_16x16x4_f32(false, a, false, b, (short)0, c0, false, false);
            b.x = sW[r0 + 16 + l]; b.y = sW[r1 + 16 + l];
            c1 = __builtin_amdgcn_wmma_f32_16x16x4_f32(false, a, false, b, (short)0, c1, false, false);
            b.x = sW[r0 + 32 + l]; b.y = sW[r1 + 32 + l];
            c2 = __builtin_amdgcn_wmma_f32_16x16x4_f32(false, a, false, b, (short)0, c2, false, false);
            b.x = sW[r0 + 48 + l]; b.y = sW[r1 + 48 + l];
            c3 = __builtin_amdgcn_wmma_f32_16x16x4_f32(false, a, false, b, (short)0, c3, false, false);
        }

        // out is a 256MB write-once stream: non-temporal stores keep x resident in L2
#pragma unroll
        for (int r = 0; r < 8; ++r) {
            int m = r + 8 * half;
            float* o = out + (size_t)(rowsBase + rbase + m) * UDIM;
            __builtin_nontemporal_store(c0[r], o + 0  + l);
            __builtin_nontemporal_store(c1[r], o + 16 + l);
            __builtin_nontemporal_store(c2[r], o + 32 + l);
            __builtin_nontemporal_store(c3[r], o + 48 + l);
        }
    }
}

// ---------------------------------------------------------------------------
extern "C" void kernel_launch(void* const* d_in, const int* in_sizes, int n_in,
                              void* d_out, int out_size, void* d_ws, size_t ws_size,
                              hipStream_t stream) {
    const float* x = (const float*)d_in[0];
    float* out = (float*)d_out;

    const int N = in_sizes[0] / UDIM;           // 1,000,000

    float* gram = (float*)d_ws;                 // 64*64 floats
    float* W    = gram + UDIM * UDIM;           // 64*64 floats

    zero_kernel<<<(UDIM * UDIM + 255) / 256, 256, 0, stream>>>(gram, UDIM * UDIM);

    const int nChunks = N / 64;                 // 15625 (N divisible by 64)
    gram_kernel<<<512, 512, 0, stream>>>(x, gram, nChunks);

    factor_kernel<<<1, 64, 0, stream>>>(gram, W);

    const int nRowTiles = N / 16;               // 62500
    const int blocks = (nRowTiles + 7) / 8;     // 7813
    apply_kernel<<<blocks, 256, 0, stream>>>(x, W, out, nRowTiles);
}